// GraphEncoder_68693706932882
// MI455X (gfx1250) — compile-verified
//
#include <hip/hip_runtime.h>

// GraphEncoder for MI455X / gfx1250 (wave32, WMMA f32 16x16x4).
// Memory-bound problem (~70MB traffic, ~1.3 GFLOP) -> keep fp32 end-to-end,
// use V_WMMA_F32_16X16X4_F32 for the two GEMMs, rely on 192MB L2 residency.

typedef __attribute__((ext_vector_type(2))) float v2f;
typedef __attribute__((ext_vector_type(4))) float v4f;
typedef __attribute__((ext_vector_type(8))) float v8f;

#define B_ 64
#define N_ 512
#define F_ 128
#define D_ 128
#define K_ 20
#define EPS_ 1e-5f
#define NEG_SLOPE_ 0.2f

// workspace layout (float units) ------------------------------------------
#define WS_WN    0                         // 512*128   normalized embed rows
#define WS_WNT   (WS_WN   + N_*D_)         // 128*512   transposed
#define WS_COS   (WS_WNT  + D_*N_)         // 512*512   cosine matrix
#define WS_LWT   (WS_COS  + N_*N_)         // 128*128   lin_w transposed (F x D)
#define WS_XLIN  (WS_LWT  + F_*D_)         // 64*512*128
#define WS_SI    (WS_XLIN + B_*N_*D_)      // 64*512
#define WS_SJ    (WS_SI   + B_*N_)         // 64*512
#define WS_AGG   (WS_SJ   + B_*N_)         // 64*512*128
#define WS_SUM   (WS_AGG  + B_*N_*D_)      // 128
#define WS_SQ    (WS_SUM  + D_)            // 128
#define WS_TOPK  (WS_SQ   + D_)            // 512*20 ints
// total ~34 MB

// --------------------------------------------------------------------------
// 1. row-normalize embed_w -> wn (N x D) and wnT (D x N)
__global__ __launch_bounds__(128) void k_norm_embed(const float* __restrict__ w,
                                                    float* __restrict__ wn,
                                                    float* __restrict__ wnT) {
    int n = blockIdx.x, d = threadIdx.x;
    float v = w[n * D_ + d];
    float s = v * v;
    #pragma unroll
    for (int m = 16; m >= 1; m >>= 1) s += __shfl_xor(s, m, 32);
    __shared__ float p[4];
    if ((d & 31) == 0) p[d >> 5] = s;
    __syncthreads();
    float tot = p[0] + p[1] + p[2] + p[3];
    float r = rsqrtf(tot);
    float o = v * r;
    wn[n * D_ + d]  = o;
    wnT[d * N_ + n] = o;
}

// 2. transpose lin_w (D x F) -> lin_wT (F x D); zero BN accumulators
__global__ __launch_bounds__(256) void k_prep(const float* __restrict__ lin_w,
                                              float* __restrict__ lwT,
                                              float* __restrict__ sums,
                                              float* __restrict__ sqs) {
    int i = blockIdx.x * 256 + threadIdx.x;
    if (i < F_ * D_) {
        int dd = i / F_, f = i % F_;
        lwT[f * D_ + dd] = lin_w[dd * F_ + f];
    }
    if (i < D_) { sums[i] = 0.f; sqs[i] = 0.f; }
}

// --------------------------------------------------------------------------
// WMMA fragment conventions (ISA 7.12.2, f32):
//  A 16x4 : lane = h*16+m (h=lane/16), vgpr j holds A[m][2h+j]
//  B 4x16 : lane = h*16+n,             vgpr j holds B[2h+j][n]
//  C 16x16: lane = h*16+n,             vgpr r holds C[r+8h][n]

// 3. cos = wn (N x D) @ wnT (D x N).  1 wave = 16 rows x 128 cols strip.
__global__ __launch_bounds__(128) void k_cos_wmma(const float* __restrict__ wn,
                                                  const float* __restrict__ wnT,
                                                  float* __restrict__ cosm) {
    int lane = threadIdx.x & 31, wave = threadIdx.x >> 5;
    int h = lane >> 4, ln = lane & 15;
    int mgrp = blockIdx.x & 7;        // 8 * 4 waves = 32 row-strips
    int cgrp = blockIdx.x >> 3;       // 4 col groups of 128
    int m0 = (mgrp * 4 + wave) * 16;
    int n0 = cgrp * 128;
    int m  = m0 + ln;

    v8f acc[8] = {};
    for (int k0 = 0; k0 < D_; k0 += 4) {
        v2f a = *(const v2f*)(wn + m * D_ + k0 + 2 * h);   // contiguous pair
        #pragma unroll
        for (int t = 0; t < 8; ++t) {
            v2f bf;
            bf.x = wnT[(k0 + 2 * h + 0) * N_ + n0 + t * 16 + ln];
            bf.y = wnT[(k0 + 2 * h + 1) * N_ + n0 + t * 16 + ln];
            acc[t] = __builtin_amdgcn_wmma_f32_16x16x4_f32(
                false, a, false, bf, (short)0, acc[t], false, false);
        }
    }
    #pragma unroll
    for (int t = 0; t < 8; ++t)
        #pragma unroll
        for (int r = 0; r < 8; ++r)
            cosm[(m0 + r + 8 * h) * N_ + n0 + t * 16 + ln] = acc[t][r];
}

// 4. per-row top-K (descending, ties -> lower index, matching jax.lax.top_k)
__global__ __launch_bounds__(32) void k_topk(const float* __restrict__ cosm,
                                             int* __restrict__ topk) {
    int row = blockIdx.x, lane = threadIdx.x;
    float v[16];
    #pragma unroll
    for (int t = 0; t < 16; ++t) v[t] = cosm[row * N_ + lane + 32 * t];
    unsigned sel = 0;
    for (int k = 0; k < K_; ++k) {
        float bv = -__builtin_inff();
        int   bi = N_;
        #pragma unroll
        for (int t = 0; t < 16; ++t) {
            if (!((sel >> t) & 1)) {
                int idx = lane + 32 * t;
                if (v[t] > bv || (v[t] == bv && idx < bi)) { bv = v[t]; bi = idx; }
            }
        }
        #pragma unroll
        for (int m = 16; m >= 1; m >>= 1) {
            float ov = __shfl_xor(bv, m, 32);
            int   oi = __shfl_xor(bi, m, 32);
            if (ov > bv || (ov == bv && oi < bi)) { bv = ov; bi = oi; }
        }
        if (lane == (bi & 31)) sel |= 1u << (bi >> 5);
        if (lane == 0) topk[row * K_ + k] = bi;
    }
}

// 5. x_lin[b] = data[b]^T (N x F) @ lin_wT (F x D).  1 wave = 16 x 128 strip.
__global__ __launch_bounds__(128) void k_xlin_wmma(const float* __restrict__ data,
                                                   const float* __restrict__ lwT,
                                                   float* __restrict__ xlin) {
    int lane = threadIdx.x & 31, wave = threadIdx.x >> 5;
    int h = lane >> 4, ln = lane & 15;
    int b    = blockIdx.x >> 3;
    int mgrp = blockIdx.x & 7;
    int m0 = (mgrp * 4 + wave) * 16;
    int m  = m0 + ln;
    const float* db = data + (size_t)b * F_ * N_;   // (F, N) row-major

    v8f acc[8] = {};
    for (int k0 = 0; k0 < F_; k0 += 4) {
        v2f a;
        a.x = db[(k0 + 2 * h + 0) * N_ + m];        // A[m][k] = data[b][k][m]
        a.y = db[(k0 + 2 * h + 1) * N_ + m];
        __builtin_prefetch(db + (k0 + 4) * N_ + m, 0, 0);  // global_prefetch_b8
        #pragma unroll
        for (int t = 0; t < 8; ++t) {
            v2f bf;
            bf.x = lwT[(k0 + 2 * h + 0) * D_ + t * 16 + ln];
            bf.y = lwT[(k0 + 2 * h + 1) * D_ + t * 16 + ln];
            acc[t] = __builtin_amdgcn_wmma_f32_16x16x4_f32(
                false, a, false, bf, (short)0, acc[t], false, false);
        }
    }
    float* xb = xlin + (size_t)b * N_ * D_;
    #pragma unroll
    for (int t = 0; t < 8; ++t)
        #pragma unroll
        for (int r = 0; r < 8; ++r)
            xb[(m0 + r + 8 * h) * D_ + t * 16 + ln] = acc[t][r];
}

// 6. s_i / s_j attention scalars: 1 wave per (b,n)
__global__ __launch_bounds__(256) void k_scores(const float* __restrict__ xlin,
                                                const float* __restrict__ w,
                                                const float* __restrict__ ai,
                                                const float* __restrict__ aj,
                                                const float* __restrict__ aei,
                                                const float* __restrict__ aej,
                                                float* __restrict__ si,
                                                float* __restrict__ sj) {
    int wid  = (blockIdx.x * 256 + threadIdx.x) >> 5;   // = b*N + n
    int lane = threadIdx.x & 31;
    int n = wid & (N_ - 1);
    const float* xr = xlin + (size_t)wid * D_;
    const float* wr = w + n * D_;
    float s_i = 0.f, s_j = 0.f;
    #pragma unroll
    for (int e = 0; e < 4; ++e) {
        int d = lane + 32 * e;
        s_i += xr[d] * ai[d] + wr[d] * aei[d];
        s_j += xr[d] * aj[d] + wr[d] * aej[d];
    }
    #pragma unroll
    for (int m = 16; m >= 1; m >>= 1) {
        s_i += __shfl_xor(s_i, m, 32);
        s_j += __shfl_xor(s_j, m, 32);
    }
    if (lane == 0) { si[wid] = s_i; sj[wid] = s_j; }
}

// 7. softmax(leaky_relu(s_i + s_j[idx])) gather-aggregate: 1 wave per (b,n)
__global__ __launch_bounds__(256) void k_agg(const float* __restrict__ xlin,
                                             const int* __restrict__ topk,
                                             const float* __restrict__ si,
                                             const float* __restrict__ sj,
                                             const float* __restrict__ bias,
                                             float* __restrict__ agg,
                                             float* __restrict__ att_out) {
    int wid  = (blockIdx.x * 256 + threadIdx.x) >> 5;
    int lane = threadIdx.x & 31;
    int b = wid >> 9, n = wid & (N_ - 1);

    int   idxs[K_];
    float alpha[K_];
    float s_i = si[wid];
    float mx = -__builtin_inff();
    #pragma unroll
    for (int k = 0; k < K_; ++k) {
        int idx = topk[n * K_ + k];
        idxs[k] = idx;
        float a = s_i + sj[b * N_ + idx];
        a = a > 0.f ? a : NEG_SLOPE_ * a;
        alpha[k] = a;
        mx = fmaxf(mx, a);
    }
    float sum = 0.f;
    #pragma unroll
    for (int k = 0; k < K_; ++k) { alpha[k] = __expf(alpha[k] - mx); sum += alpha[k]; }
    float inv = 1.f / sum;

    int d0 = lane * 4;
    v4f accv = *(const v4f*)(bias + d0);
    #pragma unroll
    for (int k = 0; k < K_; ++k) {
        v4f xv = *(const v4f*)(xlin + ((size_t)b * N_ + idxs[k]) * D_ + d0);
        accv += (alpha[k] * inv) * xv;
    }
    *(v4f*)(agg + (size_t)wid * D_ + d0) = accv;

    if (b == 0 && lane == 0) {
        #pragma unroll
        for (int k = 0; k < K_; ++k) att_out[n * K_ + k] = alpha[k] * inv;
    }
}

// 8. batch-norm statistics over B*N rows (coalesced in d, atomics once/block)
__global__ __launch_bounds__(256) void k_bnstats(const float* __restrict__ agg,
                                                 float* __restrict__ sums,
                                                 float* __restrict__ sqs) {
    __shared__ float s1[256], s2[256];
    int d = threadIdx.x & 127;
    int half = threadIdx.x >> 7;
    float s = 0.f, q = 0.f;
    size_t row0 = (size_t)blockIdx.x * 128;
    for (int r = half; r < 128; r += 2) {
        float v = agg[(row0 + r) * D_ + d];
        s += v; q += v * v;
    }
    s1[threadIdx.x] = s; s2[threadIdx.x] = q;
    __syncthreads();
    if (half == 0) {
        atomicAdd(&sums[d], s1[d] + s1[d + 128]);
        atomicAdd(&sqs[d],  s2[d] + s2[d + 128]);
    }
}

// 9. normalize + relu + * embed_w  -> d_out
__global__ __launch_bounds__(256) void k_final(const float* __restrict__ agg,
                                               const float* __restrict__ sums,
                                               const float* __restrict__ sqs,
                                               const float* __restrict__ gamma,
                                               const float* __restrict__ beta,
                                               const float* __restrict__ w,
                                               float* __restrict__ out) {
    size_t i = (size_t)blockIdx.x * 256 + threadIdx.x;
    int d = (int)(i & (D_ - 1));
    int n = (int)((i >> 7) & (N_ - 1));
    const float M = (float)(B_ * N_);
    float mean = sums[d] / M;
    float var  = sqs[d] / M - mean * mean;
    float v = agg[i];
    float y = (v - mean) * rsqrtf(var + EPS_) * gamma[d] + beta[d];
    y = fmaxf(y, 0.f);
    out[i] = y * w[n * D_ + d];
}

// 10. new_edge_index (int64): src = topk flat, dst = repeat(arange(N), K)
__global__ __launch_bounds__(256) void k_edges(const int* __restrict__ topk,
                                               long long* __restrict__ edges) {
    int i = blockIdx.x * 256 + threadIdx.x;
    if (i < N_ * K_) {
        edges[i]            = (long long)topk[i];
        edges[N_ * K_ + i]  = (long long)(i / K_);
    }
}

// --------------------------------------------------------------------------
extern "C" void kernel_launch(void* const* d_in, const int* in_sizes, int n_in,
                              void* d_out, int out_size, void* d_ws, size_t ws_size,
                              hipStream_t stream) {
    const float* data    = (const float*)d_in[0];
    const float* embed_w = (const float*)d_in[1];
    const float* lin_w   = (const float*)d_in[2];
    const float* att_i   = (const float*)d_in[3];
    const float* att_j   = (const float*)d_in[4];
    const float* att_em_i= (const float*)d_in[5];
    const float* att_em_j= (const float*)d_in[6];
    const float* bias    = (const float*)d_in[7];
    const float* bn_g    = (const float*)d_in[8];
    const float* bn_b    = (const float*)d_in[9];

    float* ws   = (float*)d_ws;
    float* wn   = ws + WS_WN;
    float* wnT  = ws + WS_WNT;
    float* cosm = ws + WS_COS;
    float* lwT  = ws + WS_LWT;
    float* xlin = ws + WS_XLIN;
    float* si   = ws + WS_SI;
    float* sj   = ws + WS_SJ;
    float* agg  = ws + WS_AGG;
    float* sums = ws + WS_SUM;
    float* sqs  = ws + WS_SQ;
    int*   topk = (int*)(ws + WS_TOPK);

    float* out      = (float*)d_out;                    // B*N*D fp32
    float* att_out  = out + (size_t)B_ * N_ * D_;       // N*K fp32
    long long* edges = (long long*)(att_out + N_ * K_); // 2*N*K int64 (8B aligned)

    k_norm_embed<<<N_, 128, 0, stream>>>(embed_w, wn, wnT);
    k_prep      <<<64, 256, 0, stream>>>(lin_w, lwT, sums, sqs);
    k_cos_wmma  <<<32, 128, 0, stream>>>(wn, wnT, cosm);
    k_topk      <<<N_, 32, 0, stream>>>(cosm, topk);
    k_xlin_wmma <<<B_ * 8, 128, 0, stream>>>(data, lwT, xlin);
    k_scores    <<<(B_ * N_) / 8, 256, 0, stream>>>(xlin, embed_w, att_i, att_j,
                                                    att_em_i, att_em_j, si, sj);
    k_agg       <<<(B_ * N_) / 8, 256, 0, stream>>>(xlin, topk, si, sj, bias,
                                                    agg, att_out);
    k_bnstats   <<<(B_ * N_) / 128, 256, 0, stream>>>(agg, sums, sqs);
    k_final     <<<(B_ * N_ * D_) / 256, 256, 0, stream>>>(agg, sums, sqs,
                                                           bn_g, bn_b, embed_w, out);
    k_edges     <<<(N_ * K_ + 255) / 256, 256, 0, stream>>>(topk, edges);
}